// Logistic_Regressor_30803505447552
// MI455X (gfx1250) — compile-verified
//
#include <hip/hip_runtime.h>

#define NN 100000
#define NE 3200000
#define CH 128
#define OC 64
#define NG 128

typedef __attribute__((ext_vector_type(2))) float v2f;
typedef __attribute__((ext_vector_type(8))) float v8f;

// ---------------- degree / normalization ----------------
__global__ void k_deg_init(float* __restrict__ deg) {
  int i = blockIdx.x * blockDim.x + threadIdx.x;
  if (i < NN) deg[i] = 1.0f;            // self-loop weight 1.0
}

__global__ void k_deg_edges(const int* __restrict__ col,
                            const float* __restrict__ w,
                            float* __restrict__ deg) {
  int e = blockIdx.x * blockDim.x + threadIdx.x;
  if (e < NE) atomicAdd(&deg[col[e]], w[e]);
}

__global__ void k_dinv(float* __restrict__ deg) {
  int i = blockIdx.x * blockDim.x + threadIdx.x;
  if (i < NN) {
    float d = deg[i];
    deg[i] = (d > 0.0f) ? rsqrtf(d) : 0.0f;
  }
}

// ---------------- dense GEMM: [NN,128] @ [128,128] via V_WMMA_F32_16X16X4_F32 ----------------
// 4 waves per block, one 16-row M-tile per wave. W staged once per block into LDS in a
// K-pair-interleaved layout: sW[((k>>1)*CH + n)*2 + (k&1)] = W[k][n], so each B fragment
// (two K-adjacent values at the same N) is one aligned 8-byte ds_load_b64.
__global__ __launch_bounds__(128) void k_gemm128(const float* __restrict__ A,
                                                 const float* __restrict__ W,
                                                 float* __restrict__ out) {
  __shared__ float sW[CH * CH];        // 64 KB
  const int tid = threadIdx.x;

  // stage W (coalesced float4 global reads, interleaved LDS writes)
  for (int i = tid; i < CH * CH / 4; i += 128) {
    const int k = i >> 5;              // row of W
    const int n = (i & 31) * 4;        // col of W
    float4 v = ((const float4*)W)[i];
    const int base = (k >> 1) * (CH * 2) + (k & 1);
    sW[base + (n + 0) * 2] = v.x;
    sW[base + (n + 1) * 2] = v.y;
    sW[base + (n + 2) * 2] = v.z;
    sW[base + (n + 3) * 2] = v.w;
  }
  __syncthreads();

  const int wave  = tid >> 5;
  const int lane  = tid & 31;
  const int mtile = blockIdx.x * 4 + wave;
  if (mtile >= NN / 16) return;        // 6250 tiles total

  const int mbase = mtile * 16;
  const int mrow  = lane & 15;
  const int khalf = lane >> 4;         // 0 -> K{0,1}, 1 -> K{2,3}

  v8f acc[8] = {};
  const float* arow = A + (size_t)(mbase + mrow) * CH;

#pragma unroll 4
  for (int kb = 0; kb < CH / 4; ++kb) {
    const int k0 = kb * 4 + khalf * 2;
    // A fragment (16x4 f32): lanes 0-15 (M=lane, K={0,1}); lanes 16-31 (M=lane-16, K={2,3})
    v2f a = *(const v2f*)(arow + k0);                 // 8B-aligned global_load_b64
#pragma unroll
    for (int t = 0; t < 8; ++t) {
      const int n0 = t * 16;
      // B fragment: K-pair {k0,k0+1} at column n0+mrow -> one ds_load_b64
      v2f b = *(const v2f*)&sW[((k0 >> 1) * CH + n0 + mrow) * 2];
      acc[t] = __builtin_amdgcn_wmma_f32_16x16x4_f32(
          /*neg_a=*/false, a, /*neg_b=*/false, b,
          /*c_mod=*/(short)0, acc[t], /*reuse_a=*/false, /*reuse_b=*/false);
    }
  }

  // C/D layout: VGPR r -> lanes 0-15 row (mbase+r), lanes 16-31 row (mbase+8+r); col = n0 + (lane&15)
  const int srow = mbase + khalf * 8;
#pragma unroll
  for (int r = 0; r < 8; ++r) {
    float* orow = out + (size_t)(srow + r) * CH + mrow;
#pragma unroll
    for (int t = 0; t < 8; ++t) orow[t * 16] = acc[t][r];
  }
}

// ---------------- message passing ----------------
// Y[i] = dinv[i]^2 * T[i]   (self-loop term, also initializes accumulator)
__global__ void k_self_init(const float* __restrict__ T,
                            const float* __restrict__ dinv,
                            float* __restrict__ Y) {
  int idx = blockIdx.x * blockDim.x + threadIdx.x;   // over NN*CH/4 float4s
  if (idx >= NN * (CH / 4)) return;
  int node = idx >> 5;                                // CH/4 == 32
  float d = dinv[node];
  float s = d * d;
  float4 t = ((const float4*)T)[idx];
  float4 y; y.x = s * t.x; y.y = s * t.y; y.z = s * t.z; y.w = s * t.w;
  ((float4*)Y)[idx] = y;
}

// wave-per-edge scatter: Y[col] += dinv[row]*w*dinv[col] * T[row]
__global__ __launch_bounds__(256) void k_scatter(const int* __restrict__ row,
                                                 const int* __restrict__ col,
                                                 const float* __restrict__ w,
                                                 const float* __restrict__ dinv,
                                                 const float* __restrict__ T,
                                                 float* __restrict__ Y) {
  int gid = blockIdx.x * blockDim.x + threadIdx.x;
  int e = gid >> 5;
  int lane = gid & 31;
  if (e >= NE) return;
  int j = row[e];
  int i = col[e];
  float norm = dinv[j] * w[e] * dinv[i];
  float4 v = ((const float4*)(T + (size_t)j * CH))[lane];  // 4 channels per lane
  float* dst = Y + (size_t)i * CH + lane * 4;
  atomicAdd(dst + 0, norm * v.x);
  atomicAdd(dst + 1, norm * v.y);
  atomicAdd(dst + 2, norm * v.z);
  atomicAdd(dst + 3, norm * v.w);
}

// H = relu(Y + b)
__global__ void k_bias_relu(const float* __restrict__ Y,
                            const float* __restrict__ b,
                            float* __restrict__ H) {
  int idx = blockIdx.x * blockDim.x + threadIdx.x;   // NN*CH
  if (idx >= NN * CH) return;
  int c = idx & (CH - 1);
  H[idx] = fmaxf(Y[idx] + b[c], 0.0f);
}

// ---------------- pooling + classifier ----------------
__global__ void k_pool_init(float* __restrict__ pooled) {
  int i = blockIdx.x * blockDim.x + threadIdx.x;
  if (i < NG * CH) pooled[i] = 0.0f;
}

__global__ __launch_bounds__(256) void k_pool(const float* __restrict__ H,
                                              const int* __restrict__ batch,
                                              float* __restrict__ pooled) {
  int gid = blockIdx.x * blockDim.x + threadIdx.x;
  int n = gid >> 5;
  int lane = gid & 31;
  if (n >= NN) return;
  int g = batch[n];
  float4 v = ((const float4*)(H + (size_t)n * CH))[lane];
  float* dst = pooled + (size_t)g * CH + lane * 4;
  atomicAdd(dst + 0, v.x);
  atomicAdd(dst + 1, v.y);
  atomicAdd(dst + 2, v.z);
  atomicAdd(dst + 3, v.w);
}

__global__ void k_classifier(const float* __restrict__ pooled,
                             const float* __restrict__ Wc,
                             const float* __restrict__ bc,
                             float* __restrict__ out) {
  int idx = blockIdx.x * blockDim.x + threadIdx.x;   // NG*OC = 8192
  if (idx >= NG * OC) return;
  int g = idx / OC;
  int o = idx - g * OC;
  float s = bc[o];
  const float* p = pooled + (size_t)g * CH;
#pragma unroll 4
  for (int c = 0; c < CH; ++c) s = fmaf(p[c], Wc[(size_t)c * OC + o], s);
  out[idx] = 1.0f / (1.0f + __expf(-s));
}

// ---------------- host-side launch ----------------
extern "C" void kernel_launch(void* const* d_in, const int* in_sizes, int n_in,
                              void* d_out, int out_size, void* d_ws, size_t ws_size,
                              hipStream_t stream) {
  (void)in_sizes; (void)n_in; (void)out_size; (void)ws_size;
  const float* x   = (const float*)d_in[0];       // [NN,128]
  const int*   ei  = (const int*)d_in[1];         // [2,NE]
  const float* ew  = (const float*)d_in[2];       // [NE]
  const int*   bat = (const int*)d_in[3];         // [NN]
  const float* W1  = (const float*)d_in[4];
  const float* b1  = (const float*)d_in[5];
  const float* W2  = (const float*)d_in[6];
  const float* b2  = (const float*)d_in[7];
  const float* Wc  = (const float*)d_in[8];
  const float* bc  = (const float*)d_in[9];
  float* out = (float*)d_out;

  const int* erow = ei;            // sources j
  const int* ecol = ei + NE;       // targets i

  float* ws   = (float*)d_ws;
  float* dinv = ws;                                   // NN floats
  float* bufA = ws + 100352;                          // NN*CH (16B aligned)
  float* bufB = bufA + (size_t)NN * CH;               // NN*CH
  float* pool = bufB + (size_t)NN * CH;               // NG*CH

  const int B = 256;
  const int GEMM_BLOCKS = (NN / 16 + 3) / 4;          // 4 waves (M-tiles) per block

  // normalization (shared by both layers)
  k_deg_init<<<(NN + B - 1) / B, B, 0, stream>>>(dinv);
  k_deg_edges<<<(NE + B - 1) / B, B, 0, stream>>>(ecol, ew, dinv);
  k_dinv<<<(NN + B - 1) / B, B, 0, stream>>>(dinv);

  // layer 1: T1 = x@W1 (bufA); Y1 (bufB); H1 = relu(Y1+b1) -> bufA
  k_gemm128<<<GEMM_BLOCKS, 128, 0, stream>>>(x, W1, bufA);
  k_self_init<<<(NN * (CH / 4) + B - 1) / B, B, 0, stream>>>(bufA, dinv, bufB);
  k_scatter<<<(int)(((size_t)NE * 32 + B - 1) / B), B, 0, stream>>>(erow, ecol, ew, dinv, bufA, bufB);
  k_bias_relu<<<(NN * CH + B - 1) / B, B, 0, stream>>>(bufB, b1, bufA);

  // layer 2: T2 = H1@W2 (bufB); Y2 (bufA); H2 = relu(Y2+b2) -> bufB
  k_gemm128<<<GEMM_BLOCKS, 128, 0, stream>>>(bufA, W2, bufB);
  k_self_init<<<(NN * (CH / 4) + B - 1) / B, B, 0, stream>>>(bufB, dinv, bufA);
  k_scatter<<<(int)(((size_t)NE * 32 + B - 1) / B), B, 0, stream>>>(erow, ecol, ew, dinv, bufB, bufA);
  k_bias_relu<<<(NN * CH + B - 1) / B, B, 0, stream>>>(bufA, b2, bufB);

  // pooling + classifier
  k_pool_init<<<(NG * CH + B - 1) / B, B, 0, stream>>>(pool);
  k_pool<<<(int)(((size_t)NN * 32 + B - 1) / B), B, 0, stream>>>(bufB, bat, pool);
  k_classifier<<<(NG * OC + B - 1) / B, B, 0, stream>>>(pool, Wc, bc, out);
}